// RateEncoder_32701880992462
// MI455X (gfx1250) — compile-verified
//
#include <hip/hip_runtime.h>

typedef __attribute__((ext_vector_type(2))) float v2f;
typedef __attribute__((ext_vector_type(8))) float v8f;

#define WAVES_PER_BLOCK 8
#define TSTEPS 32

// ---------------------------------------------------------------------------
// WMMA path: one wave32 per 16x16 output tile of cur = x @ W^T.
// K is compile-time and a multiple of 4 -> fully unrolled, branch-free WMMA
// preamble (EXEC stays all-ones), then per-lane LIF recurrence on the 8
// accumulator registers, which map to (row = row0 + r + 8*half, col = l16).
// ---------------------------------------------------------------------------
template <int K>
__global__ __launch_bounds__(256) void snn_wmma_kernel(
    const float* __restrict__ x,       // (E, K)
    const float* __restrict__ mem_in,  // (E, H)
    const float* __restrict__ W,       // (H, K)
    const float* __restrict__ bias,    // (H)
    const float* __restrict__ pbeta,   // scalar
    const float* __restrict__ pthr,    // scalar
    float* __restrict__ out,           // rate (E*H) then mem (E*H)
    int E, int H)
{
    static_assert(K % 4 == 0, "K must be a multiple of 4 for the WMMA path");

    const int lane = threadIdx.x & 31;
    const int half = lane >> 4;
    const int l16  = lane & 15;
    const int wave = threadIdx.x >> 5;

    const int  tiles_n = H >> 4;
    const long tile    = (long)blockIdx.x * WAVES_PER_BLOCK + wave;
    const int  tile_m  = (int)(tile / tiles_n);
    const int  tile_n  = (int)(tile % tiles_n);
    const int  row0    = tile_m << 4;
    const int  col0    = tile_n << 4;
    if (row0 >= E) return;             // uniform per wave

    // A fragment (16x4 f32): lane -> row = row0 + l16, K = k0 + 2*half + {0,1}
    // B fragment (4x16 f32): lane -> col = col0 + l16, same K striping,
    //   with B[k][n] = W[n][k] (cur = x @ W^T).
    const float* __restrict__ xrow = x + (long)(row0 + l16) * K + 2 * half;
    const float* __restrict__ wrow = W + (long)(col0 + l16) * K + 2 * half;

    v8f acc = {};
    #pragma unroll
    for (int k0 = 0; k0 < K; k0 += 4) {
        const v2f a = *(const v2f*)(xrow + k0);   // 8B-aligned global_load_b64
        const v2f b = *(const v2f*)(wrow + k0);
        // 8 args: (neg_a, A, neg_b, B, c_mod, C, reuse_a, reuse_b)
        acc = __builtin_amdgcn_wmma_f32_16x16x4_f32(
            false, a, false, b, (short)0, acc, false, false);
    }

    // ---- LIF recurrence (subtract reset, detached; spike = mem - thr > 0) ----
    const float bval = bias[col0 + l16];
    float beta = *pbeta;
    beta = fminf(fmaxf(beta, 0.0f), 1.0f);   // snntorch clamps learnable beta
    const float thr = *pthr;
    const long HE = (long)E * H;

    #pragma unroll
    for (int r = 0; r < 8; ++r) {
        const int  row = row0 + r + 8 * half;
        const long idx = (long)row * H + col0 + l16;
        const float cur = acc[r] + bval;

        float m = __builtin_nontemporal_load(mem_in + idx);  // read-once: NT
        float s = 0.0f;
        #pragma unroll
        for (int t = 0; t < TSTEPS; ++t) {
            const float reset = (m > thr) ? thr : 0.0f;
            m = __builtin_fmaf(beta, m, cur) - reset;
            s += (m > thr) ? 1.0f : 0.0f;
        }
        // write-once streaming stores (don't churn L2)
        __builtin_nontemporal_store(fmaxf(s * (1.0f / TSTEPS), 0.05f), out + idx);
        __builtin_nontemporal_store(m, out + HE + idx);
    }
}

// ---------------------------------------------------------------------------
// Generic fallback (arbitrary K): one thread per output element. Not used for
// the harness shapes (K = 12 takes the WMMA path), kept for safety.
// ---------------------------------------------------------------------------
__global__ __launch_bounds__(256) void snn_scalar_kernel(
    const float* __restrict__ x, const float* __restrict__ mem_in,
    const float* __restrict__ W, const float* __restrict__ bias,
    const float* __restrict__ pbeta, const float* __restrict__ pthr,
    float* __restrict__ out, int E, int H, int K)
{
    const long gid = (long)blockIdx.x * blockDim.x + threadIdx.x;
    const long HE  = (long)E * H;
    if (gid >= HE) return;
    const int row = (int)(gid / H);
    const int col = (int)(gid % H);

    float cur = bias[col];
    for (int k = 0; k < K; ++k)
        cur = __builtin_fmaf(x[(long)row * K + k], W[(long)col * K + k], cur);

    float beta = *pbeta;
    beta = fminf(fmaxf(beta, 0.0f), 1.0f);
    const float thr = *pthr;

    float m = mem_in[gid];
    float s = 0.0f;
    #pragma unroll
    for (int t = 0; t < TSTEPS; ++t) {
        const float reset = (m > thr) ? thr : 0.0f;
        m = __builtin_fmaf(beta, m, cur) - reset;
        s += (m > thr) ? 1.0f : 0.0f;
    }
    out[gid]      = fmaxf(s * (1.0f / TSTEPS), 0.05f);
    out[HE + gid] = m;
}

extern "C" void kernel_launch(void* const* d_in, const int* in_sizes, int n_in,
                              void* d_out, int out_size, void* d_ws, size_t ws_size,
                              hipStream_t stream) {
    // setup_inputs order: x, mem, W, b, beta, threshold
    const float* x    = (const float*)d_in[0];
    const float* mem  = (const float*)d_in[1];
    const float* W    = (const float*)d_in[2];
    const float* b    = (const float*)d_in[3];
    const float* beta = (const float*)d_in[4];
    const float* thr  = (const float*)d_in[5];
    float* out = (float*)d_out;

    const int H = in_sizes[3];              // 1024
    const int E = in_sizes[1] / H;          // 8192
    const int K = in_sizes[0] / E;          // 12

    const bool tiled = (E % 16 == 0) && (H % 16 == 0);
    const int  tiles  = (E >> 4) * (H >> 4);
    const int  blocks = (tiles + WAVES_PER_BLOCK - 1) / WAVES_PER_BLOCK;

    #define LAUNCH_WMMA(KK)                                                    \
        snn_wmma_kernel<KK><<<blocks, 32 * WAVES_PER_BLOCK, 0, stream>>>(      \
            x, mem, W, b, beta, thr, out, E, H)

    if (tiled) {
        switch (K) {
            case 4:  LAUNCH_WMMA(4);  return;
            case 8:  LAUNCH_WMMA(8);  return;
            case 12: LAUNCH_WMMA(12); return;   // harness shape
            case 16: LAUNCH_WMMA(16); return;
            case 20: LAUNCH_WMMA(20); return;
            case 24: LAUNCH_WMMA(24); return;
            case 32: LAUNCH_WMMA(32); return;
            default: break;
        }
    }
    #undef LAUNCH_WMMA

    const long total = (long)E * H;
    const int  fblocks = (int)((total + 255) / 256);
    snn_scalar_kernel<<<fblocks, 256, 0, stream>>>(x, mem, W, b, beta, thr, out, E, H, K);
}